// Renderer_11269994184716
// MI455X (gfx1250) — compile-verified
//
#include <hip/hip_runtime.h>
#include <stdint.h>

#define N_RAYS  8192
#define MAX_PTS 1024
#define EPSF    1e-10f
#define WPB     8            // waves per block (wave32)
#define BLOCK   (WPB * 32)
#define DEPTH   4            // async pipeline depth (chunks in flight)
#define BUFSZ   160          // floats per wave-buffer: 32 sigma + 32 dt + 96 radiance

// ---- CDNA5 async global->LDS copy (per-lane, ASYNCcnt-tracked) ----
__device__ __forceinline__ void async_copy_b32(unsigned lds_addr, int gbyteoff, const void* sbase) {
    // GVS addressing: each lane copies a dword from sbase plus its byte offset in gbyteoff to the LDS address in lds_addr
    asm volatile("global_load_async_to_lds_b32 %0, %1, %2"
                 :: "v"(lds_addr), "v"(gbyteoff), "s"(sbase)
                 : "memory");
}
__device__ __forceinline__ void wait_async_le15() { asm volatile("s_wait_asynccnt 15" ::: "memory"); }
__device__ __forceinline__ void wait_async_le10() { asm volatile("s_wait_asynccnt 10" ::: "memory"); }
__device__ __forceinline__ void wait_async_le5()  { asm volatile("s_wait_asynccnt 5"  ::: "memory"); }
__device__ __forceinline__ void wait_async_0()    { asm volatile("s_wait_asynccnt 0"  ::: "memory"); }

__global__ void __launch_bounds__(BLOCK) render_rays(
    const float* __restrict__ sigma,
    const float* __restrict__ radiance,
    const float* __restrict__ dt,
    const int*   __restrict__ numsteps,
    const float* __restrict__ bkg,
    float* __restrict__ out_rgb,
    float* __restrict__ out_mask,
    int ntotal)
{
    // Per wave: DEPTH ring buffers of [0,32)=sigma, [32,64)=dt, [64,160)=radiance chunk (96 dwords)
    __shared__ float buf[WPB][DEPTH][BUFSZ];

    const int wv   = threadIdx.x >> 5;
    const int lane = threadIdx.x & 31;
    const int ray  = blockIdx.x * WPB + wv;
    if (ray >= N_RAYS) return;

    const int count  = numsteps[2 * ray + 0];
    const int offset = numsteps[2 * ray + 1];
    const int maxidx = ntotal - 1;
    const int maxq   = 3 * ntotal - 1;

    unsigned lbase[DEPTH];
#pragma unroll
    for (int b = 0; b < DEPTH; ++b)
        lbase[b] = (unsigned)(uintptr_t)(&buf[wv][b][0]);   // low 32 bits == LDS offset

    auto issue = [&](int chunk) {
        const int      bsel = chunk & (DEPTH - 1);
        const unsigned lb   = lbase[bsel] + 4u * (unsigned)lane;
        const int      base = chunk << 5;

        int idx = offset + base + lane;                     // sigma/dt element
        if (idx > maxidx) idx = maxidx;
        const int so = idx * 4;

        int q0 = (offset + base) * 3 + lane;                // radiance chunk dwords, contiguous
        int q1 = q0 + 32;
        int q2 = q0 + 64;
        if (q0 > maxq) q0 = maxq;
        if (q1 > maxq) q1 = maxq;
        if (q2 > maxq) q2 = maxq;

        async_copy_b32(lb,         so,     sigma);          // each op spans one 128B line
        async_copy_b32(lb + 128u,  so,     dt);
        async_copy_b32(lb + 256u,  q0 * 4, radiance);
        async_copy_b32(lb + 384u,  q1 * 4, radiance);
        async_copy_b32(lb + 512u,  q2 * 4, radiance);
    };

    const int nchunks = (count + 31) >> 5;
    issue(0);
    if (1 < nchunks) issue(1);
    if (2 < nchunks) issue(2);

    float T = 1.0f, Tbkg = 1.0f;
    bool  haveTbkg = false;
    float accR = 0.f, accG = 0.f, accB = 0.f, accW = 0.f;

    for (int k = 0; k < nchunks; ++k) {
        if (k + 3 < nchunks) {
            issue(k + 3);
            wait_async_le15();                 // retires chunk k (async loads complete in order)
        } else {
            const int rem = nchunks - 1 - k;   // chunks still in flight beyond k
            if      (rem == 2) wait_async_le10();
            else if (rem == 1) wait_async_le5();
            else               wait_async_0();
        }

        const int    base = k << 5;
        const int    bsel = k & (DEPTH - 1);
        const float* wb   = &buf[wv][bsel][0];

        float s  = wb[lane];
        float d  = wb[32 + lane];
        float r0 = wb[64 + 3 * lane + 0];      // 3L mod 64 conflict-free on 64 banks
        float r1 = wb[64 + 3 * lane + 1];
        float r2 = wb[64 + 3 * lane + 2];

        const bool valid = (base + lane) < count;
        float a = valid ? (1.0f - __expf(-fmaxf(s, 0.0f) * d)) : 0.0f;
        float f = valid ? (1.0f - a + EPSF) : 1.0f;   // padded factor rounds to exactly 1.0f

        // wave32 inclusive prefix product of transmittance factors
        float p = f;
#pragma unroll
        for (int o = 1; o < 32; o <<= 1) {
            float u = __shfl_up(p, o, 32);
            if (lane >= o) p *= u;
        }
        float excl = __shfl_up(p, 1, 32);
        if (lane == 0) excl = 1.0f;
        float tot = __shfl(p, 31, 32);

        float w = a * T * excl;                 // a == 0 masks invalid lanes
        accR += w * r0; accG += w * r1; accB += w * r2; accW += w;

        // reference's trans_shift[:, -1] excludes point 1023's factor when count == 1024
        if (count == MAX_PTS && base == MAX_PTS - 32) {
            float e31 = __shfl(excl, 31, 32);
            Tbkg = T * e31;
            haveTbkg = true;
        }
        T *= tot;
    }
    if (!haveTbkg) Tbkg = T;

    // wave reduction of lane-local accumulators
#pragma unroll
    for (int o = 16; o > 0; o >>= 1) {
        accR += __shfl_xor(accR, o, 32);
        accG += __shfl_xor(accG, o, 32);
        accB += __shfl_xor(accB, o, 32);
        accW += __shfl_xor(accW, o, 32);
    }

    if (lane == 0) {
        out_rgb[3 * ray + 0] = accR + Tbkg * bkg[0];
        out_rgb[3 * ray + 1] = accG + Tbkg * bkg[1];
        out_rgb[3 * ray + 2] = accB + Tbkg * bkg[2];
        out_mask[ray]        = accW;
    }
}

extern "C" void kernel_launch(void* const* d_in, const int* in_sizes, int n_in,
                              void* d_out, int out_size, void* d_ws, size_t ws_size,
                              hipStream_t stream) {
    const float* sigma    = (const float*)d_in[0];
    const float* radiance = (const float*)d_in[1];
    const float* dt       = (const float*)d_in[2];
    const int*   numsteps = (const int*)d_in[3];
    const float* bkg      = (const float*)d_in[4];

    float* out_rgb  = (float*)d_out;                 // (8192, 3) flat
    float* out_mask = (float*)d_out + N_RAYS * 3;    // (8192,)

    const int ntotal = in_sizes[0];                  // N_RAYS * MAX_PTS
    dim3 grid(N_RAYS / WPB);
    render_rays<<<grid, BLOCK, 0, stream>>>(sigma, radiance, dt, numsteps, bkg,
                                            out_rgb, out_mask, ntotal);
}